// DecoupledEmbeddingModel_68667937129222
// MI455X (gfx1250) — compile-verified
//
#include <hip/hip_runtime.h>
#include <math.h>

typedef __attribute__((ext_vector_type(2))) float v2f;
typedef __attribute__((ext_vector_type(8))) float v8f;
typedef __attribute__((ext_vector_type(4))) int   v4i;

#define TM 128
#define TN 64
#define TK 32

// epilogue modes
#define EP_NONE      0
#define EP_RELU      1
#define EP_RNEW      2   // out = clip(0.999*aux + 0.002*(acc*scale), -3, 3)
#define EP_RESIDSUB  3   // out = aux - acc

// ---- async global->LDS path (CDNA5), guarded so either way compiles ----
#if defined(__has_builtin)
#  if __has_builtin(__builtin_amdgcn_global_load_async_to_lds_b32) && \
      __has_builtin(__builtin_amdgcn_global_load_async_to_lds_b128)
#    define USE_ASYNC_LDS 1
#  endif
#endif
#ifndef USE_ASYNC_LDS
#  define USE_ASYNC_LDS 0
#endif

typedef __attribute__((address_space(1))) int  g_i32;
typedef __attribute__((address_space(3))) int  l_i32;
typedef __attribute__((address_space(1))) v4i  g_v4i;
typedef __attribute__((address_space(3))) v4i  l_v4i;

__device__ __forceinline__ void ld_lds_f32(const float* __restrict__ g, float* l)
{
#if USE_ASYNC_LDS
    __builtin_amdgcn_global_load_async_to_lds_b32((g_i32*)g, (l_i32*)l, 0, 0);
#else
    *l = *g;
#endif
}

__device__ __forceinline__ void ld_lds_f32x4(const float* __restrict__ g, float* l)
{
#if USE_ASYNC_LDS
    __builtin_amdgcn_global_load_async_to_lds_b128((g_v4i*)g, (l_v4i*)l, 0, 0);
#else
    *(float4*)l = *(const float4*)g;
#endif
}

__device__ __forceinline__ void wait_async0()
{
#if USE_ASYNC_LDS
#  if __has_builtin(__builtin_amdgcn_s_wait_asynccnt)
    __builtin_amdgcn_s_wait_asynccnt(0);
#  else
    asm volatile("s_wait_asynccnt 0x0" ::: "memory");
#  endif
#endif
}

// ---------------------------------------------------------------------------
// Generic fp32 GEMM:  out[M,N] = epilogue( A(MxK) @ W(NxK)^T + bias )
// transA: A element (m,k) lives at A[k*M + m]
// transB: W element (n,k) lives at W[k*N + n]
// V_WMMA_F32_16X16X4_F32; 8 waves/block; 128x64 block tile; each wave a
// 32x32 tile (2x2 fragments, 4 accumulators -> 1:1 ds:wmma in inner loop).
// Double-buffered LDS with async global->LDS staging (ASYNCcnt).
// Requires M % 128 == 0.
// ---------------------------------------------------------------------------
__global__ __launch_bounds__(256)
void gemm_wmma_f32(const float* __restrict__ A, const float* __restrict__ W,
                   const float* __restrict__ bias, const float* __restrict__ aux,
                   float* __restrict__ out,
                   int M, int N, int K, int transA, int transB,
                   int mode, float scale)
{
    __shared__ float At[2][TM][TK + 1];
    __shared__ float Bt[2][TN][TK + 1];

    const int t    = threadIdx.x;
    const int lane = t & 31;
    const int w    = t >> 5;
    const int wm   = w >> 1;          // 0..3 -> 32-row strip
    const int wn   = w & 1;           // 0..1 -> 32-col strip
    const int m0   = blockIdx.y * TM;
    const int n0   = blockIdx.x * TN;

    const int half = lane >> 4;       // 0 or 1
    const int l15  = lane & 15;

    const bool vecA = (!transA) && ((K & 3) == 0);
    const bool vecB = (!transB) && ((K & 3) == 0);

    auto stage = [&](int k0, int buf) {
        // ---------------- A tile: TM x TK ----------------
        if (vecA) {
#pragma unroll
            for (int i = 0; i < (TM * TK) / (4 * 256); ++i) {   // 4 quads/thread
                int q  = i * 256 + t;
                int m  = q >> 3;
                int kq = (q & 7) * 4;
                int gm = m0 + m, gk = k0 + kq;
                if (gk < K) {
                    ld_lds_f32x4(&A[(size_t)gm * K + gk], &At[buf][m][kq]);
                } else {
                    At[buf][m][kq] = 0.f; At[buf][m][kq+1] = 0.f;
                    At[buf][m][kq+2] = 0.f; At[buf][m][kq+3] = 0.f;
                }
            }
        } else {
#pragma unroll
            for (int i = 0; i < (TM * TK) / 256; ++i) {         // 16 elems/thread
                int idx = i * 256 + t;
                int m, k;
                if (!transA) { m = idx >> 5;  k = idx & 31; }
                else         { m = idx & 127; k = idx >> 7; }
                int gm = m0 + m, gk = k0 + k;
                if (gk < K) {
                    const float* g = transA ? &A[(size_t)gk * M + gm]
                                            : &A[(size_t)gm * K + gk];
                    ld_lds_f32(g, &At[buf][m][k]);
                } else {
                    At[buf][m][k] = 0.f;
                }
            }
        }
        // ---------------- B tile: TN x TK ----------------
        if (vecB) {
#pragma unroll
            for (int i = 0; i < (TN * TK) / (4 * 256); ++i) {   // 2 quads/thread
                int q  = i * 256 + t;
                int n  = q >> 3;
                int kq = (q & 7) * 4;
                int gn = n0 + n, gk = k0 + kq;
                if (gn < N && gk < K) {
                    ld_lds_f32x4(&W[(size_t)gn * K + gk], &Bt[buf][n][kq]);
                } else {
                    Bt[buf][n][kq] = 0.f; Bt[buf][n][kq+1] = 0.f;
                    Bt[buf][n][kq+2] = 0.f; Bt[buf][n][kq+3] = 0.f;
                }
            }
        } else {
#pragma unroll
            for (int i = 0; i < (TN * TK) / 256; ++i) {         // 8 elems/thread
                int idx = i * 256 + t;
                int n, k;
                if (!transB) { n = idx >> 5; k = idx & 31; }
                else         { n = idx & 63; k = idx >> 6; }
                int gn = n0 + n, gk = k0 + k;
                if (gn < N && gk < K) {
                    const float* g = transB ? &W[(size_t)gk * N + gn]
                                            : &W[(size_t)gn * K + gk];
                    ld_lds_f32(g, &Bt[buf][n][k]);
                } else {
                    Bt[buf][n][k] = 0.f;
                }
            }
        }
    };

    v8f cacc[2][2];
#pragma unroll
    for (int ti = 0; ti < 2; ++ti)
#pragma unroll
        for (int tj = 0; tj < 2; ++tj)
            cacc[ti][tj] = (v8f){};

    const int nch = (K + TK - 1) / TK;
    const int mrow = wm * 32 + l15;
    const int nb   = wn * 32 + l15;

    stage(0, 0);
    for (int c = 0; c < nch; ++c) {
        int buf = c & 1;
        wait_async0();
        __syncthreads();                 // chunk c resident for all waves
        if (c + 1 < nch) stage((c + 1) * TK, buf ^ 1);

#pragma unroll
        for (int kk = 0; kk < TK; kk += 4) {
            int ko = kk + half * 2;
            v2f a0, a1, b0, b1;
            a0.x = At[buf][mrow][ko];      a0.y = At[buf][mrow][ko + 1];
            a1.x = At[buf][mrow + 16][ko]; a1.y = At[buf][mrow + 16][ko + 1];
            b0.x = Bt[buf][nb][ko];        b0.y = Bt[buf][nb][ko + 1];
            b1.x = Bt[buf][nb + 16][ko];   b1.y = Bt[buf][nb + 16][ko + 1];
            cacc[0][0] = __builtin_amdgcn_wmma_f32_16x16x4_f32(false, a0, false, b0,
                                                 (short)0, cacc[0][0], false, false);
            cacc[0][1] = __builtin_amdgcn_wmma_f32_16x16x4_f32(false, a0, false, b1,
                                                 (short)0, cacc[0][1], false, false);
            cacc[1][0] = __builtin_amdgcn_wmma_f32_16x16x4_f32(false, a1, false, b0,
                                                 (short)0, cacc[1][0], false, false);
            cacc[1][1] = __builtin_amdgcn_wmma_f32_16x16x4_f32(false, a1, false, b1,
                                                 (short)0, cacc[1][1], false, false);
        }
    }

    // ---- epilogue + store ----
#pragma unroll
    for (int ti = 0; ti < 2; ++ti) {
#pragma unroll
        for (int r = 0; r < 8; ++r) {
            int gm = m0 + wm * 32 + ti * 16 + half * 8 + r;
#pragma unroll
            for (int tj = 0; tj < 2; ++tj) {
                int gn = n0 + wn * 32 + tj * 16 + l15;
                if (gn < N) {
                    float v = cacc[ti][tj][r];
                    float res;
                    if (mode == EP_RNEW) {
                        float rn = 0.999f * aux[(size_t)gm * N + gn]
                                 + 0.002f * (v * scale);
                        res = fminf(fmaxf(rn, -3.f), 3.f);
                    } else if (mode == EP_RESIDSUB) {
                        res = aux[(size_t)gm * N + gn] - v;
                    } else {
                        float bv = bias ? bias[gn] : 0.f;
                        v += bv;
                        res = (mode == EP_RELU) ? fmaxf(v, 0.f) : v;
                    }
                    out[(size_t)gm * N + gn] = res;
                }
            }
        }
    }
}

// ---------------------------------------------------------------------------
// Row-wise helpers (1 block of 256 threads per row)
// ---------------------------------------------------------------------------
__device__ __forceinline__ float blk_sum(float v, float* red)
{
    int t = threadIdx.x;
    red[t] = v; __syncthreads();
#pragma unroll
    for (int s = 128; s > 0; s >>= 1) {
        if (t < s) red[t] += red[t + s];
        __syncthreads();
    }
    float r = red[0]; __syncthreads();
    return r;
}

// emb = l2norm(embedding[token]) , D=512
__global__ __launch_bounds__(256)
void k_embed(const int* __restrict__ tok, const float* __restrict__ table,
             float* __restrict__ out)
{
    __shared__ float red[256];
    int b = blockIdx.x;
    const float* src = table + (size_t)tok[b] * 512;
    float v0 = src[threadIdx.x];
    float v1 = src[threadIdx.x + 256];
    float ss = blk_sum(v0 * v0 + v1 * v1, red);
    float inv = 1.f / fmaxf(sqrtf(ss), 1e-12f);
    float* dst = out + (size_t)b * 512;
    dst[threadIdx.x]       = v0 * inv;
    dst[threadIdx.x + 256] = v1 * inv;
}

// out = l2norm(in) over len (len <= 512)
__global__ __launch_bounds__(256)
void k_l2norm(const float* __restrict__ in, float* __restrict__ out, int len)
{
    __shared__ float red[256];
    int b = blockIdx.x;
    const float* src = in + (size_t)b * len;
    int i0 = threadIdx.x, i1 = threadIdx.x + 256;
    float v0 = (i0 < len) ? src[i0] : 0.f;
    float v1 = (i1 < len) ? src[i1] : 0.f;
    float ss = blk_sum(v0 * v0 + v1 * v1, red);
    float inv = 1.f / fmaxf(sqrtf(ss), 1e-12f);
    float* dst = out + (size_t)b * len;
    if (i0 < len) dst[i0] = v0 * inv;
    if (i1 < len) dst[i1] = v1 * inv;
}

// h_t = LN(core_up + 0.1*temporal) over 512
__global__ __launch_bounds__(256)
void k_ht(const float* __restrict__ cu, const float* __restrict__ tp,
          const float* __restrict__ g, const float* __restrict__ bb,
          float* __restrict__ out)
{
    __shared__ float red[256];
    int b = blockIdx.x;
    size_t base = (size_t)b * 512;
    int i0 = threadIdx.x, i1 = threadIdx.x + 256;
    float v0 = cu[base + i0] + 0.1f * tp[base + i0];
    float v1 = cu[base + i1] + 0.1f * tp[base + i1];
    float s  = blk_sum(v0 + v1, red);
    float s2 = blk_sum(v0 * v0 + v1 * v1, red);
    float mean = s / 512.f;
    float var  = s2 / 512.f - mean * mean;
    float inv  = 1.f / sqrtf(var + 1e-5f);
    out[base + i0] = (v0 - mean) * inv * g[i0] + bb[i0];
    out[base + i1] = (v1 - mean) * inv * g[i1] + bb[i1];
}

// fused = LN(concat(emb, h_t, emb*h_t, emb-h_t)) over 2048
__global__ __launch_bounds__(256)
void k_fused(const float* __restrict__ emb, const float* __restrict__ ht,
             const float* __restrict__ g, const float* __restrict__ bb,
             float* __restrict__ out)
{
    __shared__ float red[256];
    int b = blockIdx.x;
    const float* e = emb + (size_t)b * 512;
    const float* h = ht  + (size_t)b * 512;
    float v[8];
    float s = 0.f, s2 = 0.f;
#pragma unroll
    for (int si = 0; si < 8; ++si) {
        int i = threadIdx.x + si * 256;
        int seg = i >> 9, j = i & 511;
        float ev = e[j], hv = h[j];
        float x = (seg == 0) ? ev : (seg == 1) ? hv : (seg == 2) ? ev * hv : ev - hv;
        v[si] = x; s += x; s2 += x * x;
    }
    s  = blk_sum(s, red);
    s2 = blk_sum(s2, red);
    float mean = s / 2048.f;
    float var  = s2 / 2048.f - mean * mean;
    float inv  = 1.f / sqrtf(var + 1e-5f);
    float* dst = out + (size_t)b * 2048;
#pragma unroll
    for (int si = 0; si < 8; ++si) {
        int i = threadIdx.x + si * 256;
        dst[i] = (v[si] - mean) * inv * g[i] + bb[i];
    }
}

// x = LN(gelu_exact(in)) over 1024
__global__ __launch_bounds__(256)
void k_gelu_ln(const float* __restrict__ in, const float* __restrict__ g,
               const float* __restrict__ bb, float* __restrict__ out)
{
    __shared__ float red[256];
    int b = blockIdx.x;
    const float* src = in + (size_t)b * 1024;
    float v[4];
    float s = 0.f, s2 = 0.f;
#pragma unroll
    for (int si = 0; si < 4; ++si) {
        int i = threadIdx.x + si * 256;
        float x = src[i];
        float gv = 0.5f * x * (1.f + erff(x * 0.70710678118654752f));
        v[si] = gv; s += gv; s2 += gv * gv;
    }
    s  = blk_sum(s, red);
    s2 = blk_sum(s2, red);
    float mean = s / 1024.f;
    float var  = s2 / 1024.f - mean * mean;
    float inv  = 1.f / sqrtf(var + 1e-5f);
    float* dst = out + (size_t)b * 1024;
#pragma unroll
    for (int si = 0; si < 4; ++si) {
        int i = threadIdx.x + si * 256;
        dst[i] = (v[si] - mean) * inv * g[i] + bb[i];
    }
}

// out = LN(x2 + emb) over 512
__global__ __launch_bounds__(256)
void k_out_ln(const float* __restrict__ x, const float* __restrict__ emb,
              const float* __restrict__ g, const float* __restrict__ bb,
              float* __restrict__ out)
{
    __shared__ float red[256];
    int b = blockIdx.x;
    size_t base = (size_t)b * 512;
    int i0 = threadIdx.x, i1 = threadIdx.x + 256;
    float v0 = x[base + i0] + emb[base + i0];
    float v1 = x[base + i1] + emb[base + i1];
    float s  = blk_sum(v0 + v1, red);
    float s2 = blk_sum(v0 * v0 + v1 * v1, red);
    float mean = s / 512.f;
    float var  = s2 / 512.f - mean * mean;
    float inv  = 1.f / sqrtf(var + 1e-5f);
    out[base + i0] = (v0 - mean) * inv * g[i0] + bb[i0];
    out[base + i1] = (v1 - mean) * inv * g[i1] + bb[i1];
}

// ---------------------------------------------------------------------------
extern "C" void kernel_launch(void* const* d_in, const int* in_sizes, int n_in,
                              void* d_out, int out_size, void* d_ws, size_t ws_size,
                              hipStream_t stream)
{
    (void)in_sizes; (void)n_in; (void)out_size; (void)ws_size;
    const int B = 2048, D = 512, C = 110, H = 1024, V = 50257;

    const int*   tok   = (const int*)  d_in[0];
    const float* hprev = (const float*)d_in[1];
    const float* Rw    = (const float*)d_in[2];
    const float* table = (const float*)d_in[3];
    const float* cw    = (const float*)d_in[4];
    const float* cb    = (const float*)d_in[5];
    const float* w1    = (const float*)d_in[6];
    const float* b1    = (const float*)d_in[7];
    const float* w2    = (const float*)d_in[8];
    const float* b2    = (const float*)d_in[9];
    const float* upw   = (const float*)d_in[10];
    const float* upb   = (const float*)d_in[11];
    const float* rng   = (const float*)d_in[12];
    const float* rnb   = (const float*)d_in[13];
    const float* fg    = (const float*)d_in[14];
    const float* fb    = (const float*)d_in[15];
    const float* p1w   = (const float*)d_in[16];
    const float* p1b   = (const float*)d_in[17];
    const float* plg   = (const float*)d_in[18];
    const float* plb   = (const float*)d_in[19];
    const float* p2w   = (const float*)d_in[20];
    const float* p2b   = (const float*)d_in[21];
    const float* og    = (const float*)d_in[22];
    const float* ob    = (const float*)d_in[23];
    float* logits = (float*)d_out;

    // workspace carve-up (floats)
    float* ws = (float*)d_ws;
    size_t o = 0;
    float* emb      = ws + o; o += (size_t)B * D;
    float* core_lin = ws + o; o += (size_t)B * C;     // reused as co_lin
    float* core_in  = ws + o; o += (size_t)B * C;
    float* h1       = ws + o; o += (size_t)B * H;     // reused as x1
    float* core_out = ws + o; o += (size_t)B * C;
    float* cu_lin   = ws + o; o += (size_t)B * D;     // reused as temporal
    float* core_up  = ws + o; o += (size_t)B * D;
    float* eps      = ws + o; o += (size_t)B * D;     // reused as x2
    float* Rnew     = ws + o; o += (size_t)D * D;
    float* h_t      = ws + o; o += (size_t)B * D;
    float* fused    = ws + o; o += (size_t)B * 4 * D;
    float* x1n      = ws + o; o += (size_t)B * H;
    float* outn     = ws + o; o += (size_t)B * D;

    dim3 blk(256);
    auto grd = [](int N_, int M_) { return dim3((N_ + TN - 1) / TN, M_ / TM); };

    // 1. emb = l2norm(gather)
    k_embed<<<B, blk, 0, stream>>>(tok, table, emb);
    // 2-3. core_in = l2norm(emb @ cw^T + cb)
    gemm_wmma_f32<<<grd(C, B), blk, 0, stream>>>(emb, cw, cb, nullptr, core_lin,
                                                 B, C, D, 0, 0, EP_NONE, 1.f);
    k_l2norm<<<B, blk, 0, stream>>>(core_lin, core_in, C);
    // 4. h1 = relu(core_in @ w1^T + b1)
    gemm_wmma_f32<<<grd(H, B), blk, 0, stream>>>(core_in, w1, b1, nullptr, h1,
                                                 B, H, C, 0, 0, EP_RELU, 1.f);
    // 5-6. core_out = l2norm(h1 @ w2^T + b2)
    gemm_wmma_f32<<<grd(C, B), blk, 0, stream>>>(h1, w2, b2, nullptr, core_lin,
                                                 B, C, H, 0, 0, EP_NONE, 1.f);
    k_l2norm<<<B, blk, 0, stream>>>(core_lin, core_out, C);
    // 7-8. core_up = l2norm(core_out @ upw^T + upb)
    gemm_wmma_f32<<<grd(D, B), blk, 0, stream>>>(core_out, upw, upb, nullptr, cu_lin,
                                                 B, D, C, 0, 0, EP_NONE, 1.f);
    k_l2norm<<<B, blk, 0, stream>>>(cu_lin, core_up, D);
    // 9. eps = core_up - h_prev @ R
    gemm_wmma_f32<<<grd(D, B), blk, 0, stream>>>(hprev, Rw, nullptr, core_up, eps,
                                                 B, D, D, 0, 1, EP_RESIDSUB, 1.f);
    // 10. R_new = clip(0.999*R + 0.002*(h_prev^T @ eps)/B)
    gemm_wmma_f32<<<grd(D, D), blk, 0, stream>>>(hprev, eps, nullptr, Rw, Rnew,
                                                 D, D, B, 1, 1, EP_RNEW, 1.f / (float)B);
    // 11. temporal = h_prev @ R_new
    gemm_wmma_f32<<<grd(D, B), blk, 0, stream>>>(hprev, Rnew, nullptr, nullptr, cu_lin,
                                                 B, D, D, 0, 1, EP_NONE, 1.f);
    // 12. h_t = LN(core_up + 0.1*temporal)
    k_ht<<<B, blk, 0, stream>>>(core_up, cu_lin, rng, rnb, h_t);
    // 13. fused = LN(concat4)
    k_fused<<<B, blk, 0, stream>>>(emb, h_t, fg, fb, fused);
    // 14-15. x1n = LN(gelu(fused @ p1^T + p1b))
    gemm_wmma_f32<<<grd(H, B), blk, 0, stream>>>(fused, p1w, p1b, nullptr, h1,
                                                 B, H, 4 * D, 0, 0, EP_NONE, 1.f);
    k_gelu_ln<<<B, blk, 0, stream>>>(h1, plg, plb, x1n);
    // 16. x2 = x1n @ p2^T + p2b
    gemm_wmma_f32<<<grd(D, B), blk, 0, stream>>>(x1n, p2w, p2b, nullptr, eps,
                                                 B, D, H, 0, 0, EP_NONE, 1.f);
    // 17. out = LN(x2 + emb)
    k_out_ln<<<B, blk, 0, stream>>>(eps, emb, og, ob, outn);
    // 18. logits = out @ embedding^T  (M=2048, N=50257, K=512; store-bound)
    gemm_wmma_f32<<<grd(V, B), blk, 0, stream>>>(outn, table, nullptr, nullptr, logits,
                                                 B, V, D, 0, 0, EP_NONE, 1.f);
}